// NP_35304631173303
// MI455X (gfx1250) — compile-verified
//
#include <hip/hip_runtime.h>

typedef __attribute__((ext_vector_type(4))) float vf4;
typedef __attribute__((ext_vector_type(2))) float v2f;
typedef __attribute__((ext_vector_type(8))) float v8f;

#define THREADS 256

// One workgroup per row. Streams the row once with NT b128 loads, keeps the
// five Pearson statistics in registers, reduces wave32-wide with xor-shuffles,
// combines the 8 waves through LDS, and writes the per-row loss to scratch.
__global__ __launch_bounds__(THREADS) void pearson_rows(
    const float* __restrict__ X, const float* __restrict__ Y,
    float* __restrict__ rowloss, int N) {
  const int row = blockIdx.x;
  const long long base = (long long)row * (long long)N;
  const vf4* __restrict__ x4 = (const vf4*)(X + base);
  const vf4* __restrict__ y4 = (const vf4*)(Y + base);
  const int nvec = N >> 2;

  float sx = 0.f, sy = 0.f, sxy = 0.f, sxx = 0.f, syy = 0.f;

#pragma unroll 4
  for (int i = threadIdx.x; i < nvec; i += THREADS) {
    vf4 a = __builtin_nontemporal_load(&x4[i]);
    vf4 b = __builtin_nontemporal_load(&y4[i]);
    sx += a.x + a.y + a.z + a.w;
    sy += b.x + b.y + b.z + b.w;
    sxy = fmaf(a.x, b.x, sxy);
    sxy = fmaf(a.y, b.y, sxy);
    sxy = fmaf(a.z, b.z, sxy);
    sxy = fmaf(a.w, b.w, sxy);
    sxx = fmaf(a.x, a.x, sxx);
    sxx = fmaf(a.y, a.y, sxx);
    sxx = fmaf(a.z, a.z, sxx);
    sxx = fmaf(a.w, a.w, sxx);
    syy = fmaf(b.x, b.x, syy);
    syy = fmaf(b.y, b.y, syy);
    syy = fmaf(b.z, b.z, syy);
    syy = fmaf(b.w, b.w, syy);
  }

  // wave32 intra-wave reduction (5 levels, NOT 6: CDNA5 is wave32 only)
#pragma unroll
  for (int off = 16; off > 0; off >>= 1) {
    sx  += __shfl_xor(sx, off, 32);
    sy  += __shfl_xor(sy, off, 32);
    sxy += __shfl_xor(sxy, off, 32);
    sxx += __shfl_xor(sxx, off, 32);
    syy += __shfl_xor(syy, off, 32);
  }

  __shared__ float red[8][5];
  const int wave = threadIdx.x >> 5;
  const int lane = threadIdx.x & 31;
  if (lane == 0) {
    red[wave][0] = sx;
    red[wave][1] = sy;
    red[wave][2] = sxy;
    red[wave][3] = sxx;
    red[wave][4] = syy;
  }
  __syncthreads();

  if (threadIdx.x == 0) {
    float tx = 0.f, ty = 0.f, txy = 0.f, txx = 0.f, tyy = 0.f;
#pragma unroll
    for (int w = 0; w < 8; ++w) {
      tx  += red[w][0];
      ty  += red[w][1];
      txy += red[w][2];
      txx += red[w][3];
      tyy += red[w][4];
    }
    const float fN = (float)N;
    const float num = fmaf(fN, txy, -(tx * ty));
    const float vx  = fmaf(fN, txx, -(tx * tx));
    const float vy  = fmaf(fN, tyy, -(ty * ty));
    const float den = sqrtf(vx * vy);
    rowloss[row] = 1.0f - num / den;
  }
}

// Single-block mean over the B row losses. After LDS folding to 64 partials,
// wave 0 finishes the sum with one V_WMMA_F32_16X16X4_F32:
//   A = ones(16x4)  ->  D[i][j] = sum_k B[k][j]   (column sums of B, 64 -> 16)
// B-matrix layout (32-bit B, 4x16, rows striped across lanes per VGPR):
//   VGPR0: lanes0-15 = K0 row, lanes16-31 = K1 row; VGPR1: K2, K3
//   => lane l supplies b.x = p[l], b.y = p[l+32].
// D layout replicates the 16 column sums in VGPR0 across all 32 lanes, so
// 4 xor-shuffles finish the 16-way sum.
__global__ __launch_bounds__(THREADS) void mean_loss_wmma(
    const float* __restrict__ rowloss, float* __restrict__ out, int B) {
  __shared__ float p[THREADS];
  float s = 0.f;
  for (int i = threadIdx.x; i < B; i += THREADS) s += rowloss[i];
  p[threadIdx.x] = s;
  __syncthreads();

  if (threadIdx.x < 64) {
    p[threadIdx.x] = p[threadIdx.x] + p[threadIdx.x + 64] +
                     p[threadIdx.x + 128] + p[threadIdx.x + 192];
  }
  __syncthreads();

  if (threadIdx.x < 32) {  // uniform branch: wave 0 has EXEC all-1s (WMMA req)
    v2f bm;
    bm.x = p[threadIdx.x];
    bm.y = p[threadIdx.x + 32];
    v2f am;
    am.x = 1.0f;
    am.y = 1.0f;
    v8f c = {};
    // 8 args: (neg_a, A, neg_b, B, c_mod, C, reuse_a, reuse_b)
    v8f d = __builtin_amdgcn_wmma_f32_16x16x4_f32(
        false, am, false, bm, (short)0, c, false, false);
    float col = d[0];  // lane l: colsum of column (l & 15)
#pragma unroll
    for (int off = 8; off > 0; off >>= 1) col += __shfl_xor(col, off, 32);
    if (threadIdx.x == 0) out[0] = col * (1.0f / (float)B);
  }
}

extern "C" void kernel_launch(void* const* d_in, const int* in_sizes, int n_in,
                              void* d_out, int out_size, void* d_ws,
                              size_t ws_size, hipStream_t stream) {
  const float* preds  = (const float*)d_in[0];
  const float* labels = (const float*)d_in[1];
  float* out = (float*)d_out;
  float* rowloss = (float*)d_ws;  // B floats of scratch

  const int N = 8192;
  const int B = in_sizes[0] / N;  // 4096

  pearson_rows<<<B, THREADS, 0, stream>>>(preds, labels, rowloss, N);
  mean_loss_wmma<<<1, THREADS, 0, stream>>>(rowloss, out, B);
}